// ConcatAttention_68719477239
// MI455X (gfx1250) — compile-verified
//
#include <hip/hip_runtime.h>
#include <hip/hip_bf16.h>
#include <math.h>

#define H 1024
#define S 2048
#define B 32

typedef float v2f __attribute__((ext_vector_type(2)));
typedef float v8f __attribute__((ext_vector_type(8)));
typedef float f4v __attribute__((ext_vector_type(4)));

// ---------------------------------------------------------------------------
// K1: dec_proj[b][n] = sum_k dec[b][k] * Wd[k][n]   (32x1024 @ 1024x1024)
// 128 (row_tile, n_tile) wave-jobs: 16 blocks x 8 waves.
// ---------------------------------------------------------------------------
__global__ void __launch_bounds__(256) dec_proj_kernel(const float* __restrict__ dec,
                                                       const float* __restrict__ Wd,
                                                       float* __restrict__ decp) {
    const int tid  = threadIdx.x;
    const int lane = tid & 31;
    const int warp = tid >> 5;
    const int half = lane >> 4;   // which 16-lane half
    const int ln   = lane & 15;
    const int wg   = blockIdx.x * 8 + warp;  // 0..127
    const int rt   = wg >> 6;                // row tile 0..1  (16 rows each)
    const int nt   = wg & 63;                // col tile 0..63
    const int m    = rt * 16 + ln;           // A row == batch index
    const int n0   = nt * 16;

    v8f c = {0.f, 0.f, 0.f, 0.f, 0.f, 0.f, 0.f, 0.f};
    for (int k0 = 0; k0 < H; k0 += 4) {
        v2f a, b;
        // A frag: a[j] = A[m][k0 + 2*half + j]
        a.x = dec[(size_t)m * H + k0 + 2 * half];
        a.y = dec[(size_t)m * H + k0 + 2 * half + 1];
        // B frag: b[j] = Wd[k0 + 2*half + j][n0 + ln]
        b.x = Wd[(size_t)(k0 + 2 * half) * H + n0 + ln];
        b.y = Wd[(size_t)(k0 + 2 * half + 1) * H + n0 + ln];
        c = __builtin_amdgcn_wmma_f32_16x16x4_f32(false, a, false, b,
                                                  (short)0, c, false, false);
    }
#pragma unroll
    for (int i = 0; i < 8; ++i) {
        const int row = rt * 16 + i + 8 * half;   // D: row = i + 8*half
        decp[(size_t)row * H + n0 + ln] = c[i];
    }
}

// ---------------------------------------------------------------------------
// K2: fused  logits[b][s] = sum_n tanh( (enc_row @ We)[n] + decp[b][n] ) * v[n]
// enc flattened [S,B,H] is a (S*B) x H row-major matrix; row r = s*B + b.
//
// BM=64 rows per block (4 WMMA row-tiles) with register blocking: each wave
// holds 4 accumulators and reuses each We B-fragment across 4 v_wmma ops
// (We L2 traffic 4 GB total vs 16 GB at BM=16); enc is read exactly once from
// HBM. The A-tile is staged with CDNA5 async global->LDS DMA (ASYNCcnt),
// bypassing VGPRs entirely; decp is folded into the WMMA C accumulator.
// ---------------------------------------------------------------------------
#define BM 64
#define A_STRIDE 1028  // 1024 + 4 pad -> 16 lanes land on distinct LDS banks

__global__ void __launch_bounds__(256) logits_kernel(const float* __restrict__ enc,
                                                     const float* __restrict__ We,
                                                     const float* __restrict__ decp,
                                                     const float* __restrict__ v,
                                                     float* __restrict__ logits) {
    __shared__ float Alds[BM * A_STRIDE];   // ~257 KB (within 320 KB WGP LDS)
    __shared__ float Lred[BM];

    const int tid  = threadIdx.x;
    const int lane = tid & 31;
    const int warp = tid >> 5;
    const int half = lane >> 4;
    const int ln   = lane & 15;
    const int r0   = blockIdx.x * BM;   // first global row of this tile

    // Stage A tile (64 rows x 1024 f32) via async global->LDS B128 DMA.
    // Flat LDS pointers truncate to the wave-relative LDS offset (ISA 10.2),
    // so the low 32 bits of &Alds[...] are the correct VDST lds address.
#if defined(__AMDGCN__)
#pragma unroll
    for (int i = 0; i < (BM * H / 4) / 256; ++i) {   // 64 xB128 per thread
        const int idx = tid + i * 256;     // float4 slot
        const int row = idx >> 8;          // 256 float4 per row
        const int c4  = idx & 255;
        unsigned lds_off = (unsigned)(uintptr_t)(&Alds[row * A_STRIDE + c4 * 4]);
        unsigned long long gaddr =
            (unsigned long long)(uintptr_t)(enc + (size_t)(r0 + row) * H + c4 * 4);
        asm volatile("global_load_async_to_lds_b128 %0, %1, off"
                     :: "v"(lds_off), "v"(gaddr) : "memory");
    }
    if (tid < BM) Lred[tid] = 0.f;
    asm volatile("s_wait_asynccnt 0x0" ::: "memory");
#else
#pragma unroll
    for (int i = 0; i < (BM * H / 4) / 256; ++i) {
        const int idx = tid + i * 256;
        const int row = idx >> 8;
        const int c4  = idx & 255;
        *(f4v*)(&Alds[row * A_STRIDE + c4 * 4]) =
            ((const f4v*)(enc + (size_t)r0 * H))[row * 256 + c4];
    }
    if (tid < BM) Lred[tid] = 0.f;
#endif
    __syncthreads();

    float logacc[4][8];
#pragma unroll
    for (int rt = 0; rt < 4; ++rt)
#pragma unroll
        for (int i = 0; i < 8; ++i) logacc[rt][i] = 0.f;

    // Each wave owns 8 of the 64 N-tiles; 4 row-tiles deep per N-tile.
    for (int t = 0; t < 8; ++t) {
        const int ncol = (warp * 8 + t) * 16 + ln;

        // Init accumulators with the broadcast decoder projection:
        // energy = tanh(decp[b] + enc_row @ We)  ->  C-init = decp.
        v8f cacc[4];
#pragma unroll
        for (int rt = 0; rt < 4; ++rt) {
#pragma unroll
            for (int i = 0; i < 8; ++i) {
                const int b = (r0 + rt * 16 + i + 8 * half) & (B - 1);
                cacc[rt][i] = decp[(size_t)b * H + ncol];
            }
        }

        const float* Abase = &Alds[ln * A_STRIDE + 2 * half];
        const float* Bp    = We + (size_t)(2 * half) * H + ncol;
        for (int k0 = 0; k0 < H; k0 += 4) {
            v2f b;
            b.x = Bp[(size_t)k0 * H];
            b.y = Bp[(size_t)k0 * H + H];
#pragma unroll
            for (int rt = 0; rt < 4; ++rt) {
                v2f a = *(const v2f*)(Abase + rt * 16 * A_STRIDE + k0); // ds_load_b64
                cacc[rt] = __builtin_amdgcn_wmma_f32_16x16x4_f32(
                    false, a, false, b, (short)0, cacc[rt], false, false);
            }
        }

        const float vn = v[ncol];
#pragma unroll
        for (int rt = 0; rt < 4; ++rt)
#pragma unroll
            for (int i = 0; i < 8; ++i)
                logacc[rt][i] += tanhf(cacc[rt][i]) * vn;
    }

    // Reduce over the 16 N-lanes within each half-wave (masks < 16 stay in half).
#pragma unroll
    for (int rt = 0; rt < 4; ++rt) {
#pragma unroll
        for (int i = 0; i < 8; ++i) {
            float x = logacc[rt][i];
            x += __shfl_xor(x, 1);
            x += __shfl_xor(x, 2);
            x += __shfl_xor(x, 4);
            x += __shfl_xor(x, 8);
            if (ln == 0)
                atomicAdd(&Lred[rt * 16 + i + 8 * half], x);  // ds_add_f32
        }
    }
    __syncthreads();

    if (tid < BM) {
        const int row = r0 + tid;
        const int b   = row & (B - 1);
        const int s   = row >> 5;
        logits[(size_t)b * S + s] = Lred[tid];
    }
}

// ---------------------------------------------------------------------------
// K3: softmax over S per batch. One block per b. S=2048, 256 threads x 8.
// ---------------------------------------------------------------------------
__global__ void __launch_bounds__(256) softmax_kernel(const float* __restrict__ logits,
                                                      float* __restrict__ scores) {
    __shared__ float red[256];
    const int b   = blockIdx.x;
    const int tid = threadIdx.x;
    const float* row = logits + (size_t)b * S;

    float m = -INFINITY;
#pragma unroll
    for (int i = 0; i < 8; ++i) m = fmaxf(m, row[tid + i * 256]);
    red[tid] = m;
    __syncthreads();
    for (int off = 128; off > 0; off >>= 1) {
        if (tid < off) red[tid] = fmaxf(red[tid], red[tid + off]);
        __syncthreads();
    }
    const float mx = red[0];
    __syncthreads();

    float e[8];
    float sum = 0.f;
#pragma unroll
    for (int i = 0; i < 8; ++i) {
        e[i] = __expf(row[tid + i * 256] - mx);
        sum += e[i];
    }
    red[tid] = sum;
    __syncthreads();
    for (int off = 128; off > 0; off >>= 1) {
        if (tid < off) red[tid] += red[tid + off];
        __syncthreads();
    }
    const float inv = 1.0f / red[0];
#pragma unroll
    for (int i = 0; i < 8; ++i)
        scores[(size_t)b * S + tid + i * 256] = e[i] * inv;
}

// ---------------------------------------------------------------------------
// K4: attn_values[b][h] = sum_s scores[b][s] * enc[s][b][h]
// Split-S (16 splits) for occupancy; merge with global f32 atomics.
// ---------------------------------------------------------------------------
__global__ void __launch_bounds__(256) zero_kernel(float* __restrict__ out, int n) {
    const int i = blockIdx.x * 256 + threadIdx.x;
    if (i < n) out[i] = 0.f;
}

__global__ void __launch_bounds__(256) weighted_sum_kernel(const float* __restrict__ enc,
                                                           const float* __restrict__ scores,
                                                           float* __restrict__ out) {
    __shared__ float sc[128];
    const int b  = blockIdx.x;               // 0..31
    const int hc = blockIdx.y;               // 0..3
    const int s0 = blockIdx.z * (S / 16);    // 128-wide S split
    const int h  = hc * 256 + threadIdx.x;

    if (threadIdx.x < 128) sc[threadIdx.x] = scores[(size_t)b * S + s0 + threadIdx.x];
    __syncthreads();

    float acc = 0.f;
    for (int i = 0; i < S / 16; ++i) {
        const int s = s0 + i;
        acc += sc[i] * enc[((size_t)s * B + b) * H + h];
    }
    atomicAdd(&out[(size_t)b * H + h], acc);  // global_atomic_add_f32
}

// ---------------------------------------------------------------------------
extern "C" void kernel_launch(void* const* d_in, const int* in_sizes, int n_in,
                              void* d_out, int out_size, void* d_ws, size_t ws_size,
                              hipStream_t stream) {
    const float* dec = (const float*)d_in[0];   // [1,B,H]
    const float* enc = (const float*)d_in[1];   // [S,B,H]
    const float* Wd  = (const float*)d_in[2];   // [H,H]
    const float* We  = (const float*)d_in[3];   // [H,H]
    const float* v   = (const float*)d_in[4];   // [H]

    float* out    = (float*)d_out;              // [B*H] values ++ [B*S] scores
    float* decp   = (float*)d_ws;               // B*H
    float* logits = decp + B * H;               // B*S
    float* scores = out + B * H;

    dec_proj_kernel<<<16, 256, 0, stream>>>(dec, Wd, decp);
    logits_kernel<<<(S * B) / BM, 256, 0, stream>>>(enc, We, decp, v, logits);
    softmax_kernel<<<B, 256, 0, stream>>>(logits, scores);
    zero_kernel<<<(B * H) / 256, 256, 0, stream>>>(out, B * H);
    weighted_sum_kernel<<<dim3(B, H / 256, 16), 256, 0, stream>>>(enc, scores, out);
}